// MRConv2d_16870631538992
// MI455X (gfx1250) — compile-verified
//
#include <hip/hip_runtime.h>
#include <math.h>

// MRConv2d (max-relative graph conv) for MI455X / gfx1250.
// B=4, C_IN=384, C_OUT=768, N=8192, K=16, GROUPS=4.
//
// Plan:
//  1) transpose x [B,C,N] -> xT [B,N,C] so neighbor gathers are coalesced
//     (all gather traffic then resides in the 192MB L2, read as full lines).
//  2) fused kernel: per (batch, 16-wide N tile) block:
//       phase 1: gather + max-relative, build interleaved merged tile
//                [2*C_IN x 16] in LDS (WMMA B-operand layout, pitch 17).
//       phase 2: grouped 192x192 GEMM with V_WMMA_F32_16X16X4_F32 (full
//                fp32, matches reference precision), bias + ReLU, store.

#define BB      4
#define C_IN    384
#define C_OUTC  768
#define NN      8192
#define KNN     16
#define NGROUP  4
#define CIN_G   192     // (2*C_IN)/GROUPS
#define COUT_G  192     // C_OUT/GROUPS
#define C2      768     // 2*C_IN
#define NT      16      // n-columns per block
#define PITCH   17      // LDS row pitch (floats) -> conflict-free phase-1 writes

typedef __attribute__((ext_vector_type(2))) float v2f;
typedef __attribute__((ext_vector_type(8))) float v8f;

// ---------------- x [B,C,N] -> xT [B,N,C] (32x32 LDS tile transpose) -------
__global__ __launch_bounds__(256) void xpose_kernel(const float* __restrict__ x,
                                                    float* __restrict__ xT) {
  __shared__ float tile[32][33];
  const int n0 = blockIdx.x * 32;       // N/32 = 256
  const int c0 = blockIdx.y * 32;       // C_IN/32 = 12
  const int b  = blockIdx.z;
  const int tx = threadIdx.x;           // 0..31
  const int ty = threadIdx.y;           // 0..7
  const float* xb = x + (size_t)b * C_IN * NN;
#pragma unroll
  for (int i = 0; i < 4; ++i) {
    const int c = c0 + ty + 8 * i;
    tile[ty + 8 * i][tx] = xb[(size_t)c * NN + n0 + tx];     // coalesced read
  }
  __syncthreads();
  float* xTb = xT + (size_t)b * NN * C_IN;
#pragma unroll
  for (int i = 0; i < 4; ++i) {
    const int n = n0 + ty + 8 * i;
    xTb[(size_t)n * C_IN + c0 + tx] = tile[tx][ty + 8 * i];  // coalesced write
  }
}

// ---------------- fused gather/max-rel + grouped WMMA GEMM -----------------
__global__ __launch_bounds__(256) void mrconv_kernel(
    const float* __restrict__ xT, const int* __restrict__ edge,
    const float* __restrict__ W, const float* __restrict__ bias,
    float* __restrict__ out) {
  // merged tile, B-operand layout: row = interleaved channel, col = local n
  __shared__ float merged[C2 * PITCH];  // 768*17*4 = 52224 B

  const int blk = blockIdx.x;               // B * N/NT = 2048 blocks
  const int b   = blk / (NN / NT);
  const int n0  = (blk % (NN / NT)) * NT;
  const int tid = threadIdx.x;

  // ---------------- phase 1: gather + max-relative ----------------
  {
    const int nl = tid >> 4;                // local n column 0..15
    const int lc = tid & 15;                // c-lane within 16-thread group
    const int n  = n0 + nl;
    const int* e0 = edge + ((size_t)b * NN + n) * KNN;          // edge_index[0]
    const int* e1 = edge + ((size_t)(BB + b) * NN + n) * KNN;   // edge_index[1]
    int i0[KNN], i1[KNN];
#pragma unroll
    for (int k = 0; k < KNN; ++k) { i0[k] = e0[k]; i1[k] = e1[k]; }

    const float* xb = xT + (size_t)b * NN * C_IN;
    const float* xc = xb + (size_t)n * C_IN;
#pragma unroll 2
    for (int ci = 0; ci < C_IN / 16; ++ci) {
      const int c = lc + 16 * ci;
      const float xv = xc[c];
      float rel = -INFINITY;
#pragma unroll
      for (int k = 0; k < KNN; ++k) {
        const float xj = xb[(size_t)i0[k] * C_IN + c];  // coalesced 64B/group
        const float xi = xb[(size_t)i1[k] * C_IN + c];
        rel = fmaxf(rel, xj - xi);
      }
      merged[(2 * c    ) * PITCH + nl] = xv;    // interleave (x_c, rel_c)
      merged[(2 * c + 1) * PITCH + nl] = rel;
    }
  }
  __syncthreads();

  // ---------------- phase 2: grouped GEMM, f32 WMMA 16x16x4 ----------------
  const int wave = tid >> 5;                // 8 waves
  const int lane = tid & 31;
  const int m15  = lane & 15;
  const int hi   = lane >> 4;               // half-wave select

  for (int t = wave; t < NGROUP * (COUT_G / 16); t += 8) {
    const int g  = t / (COUT_G / 16);
    const int ot = t % (COUT_G / 16);
    const int c0 = g * COUT_G + ot * 16;    // output-channel base of tile

    v8f acc = {};
    // A operand: lane holds W row (c0+m15), K pair = 2*hi,2*hi+1 of chunk
    const float* Wrow  = W + (size_t)(c0 + m15) * CIN_G + 2 * hi;
    const float* Bbase = merged + (size_t)(g * CIN_G) * PITCH;

#pragma unroll 4
    for (int kc = 0; kc < CIN_G / 4; ++kc) {
      v2f a;
      a[0] = Wrow[kc * 4];
      a[1] = Wrow[kc * 4 + 1];
      // B operand: VGPR0 holds rows K, K+2 (per half-wave); VGPR1 rows K+1, K+3
      const float* bp = Bbase + (size_t)(kc * 4 + 2 * hi) * PITCH + m15;
      v2f bv;
      bv[0] = bp[0];
      bv[1] = bp[PITCH];
      acc = __builtin_amdgcn_wmma_f32_16x16x4_f32(
          /*neg_a=*/false, a, /*neg_b=*/false, bv,
          /*c_mod=*/(short)0, acc, /*reuse_a=*/false, /*reuse_b=*/false);
    }

    // epilogue: bias + ReLU, store per C-layout (VGPR v -> rows v, v+8)
    float* outp = out + ((size_t)b * C_OUTC + c0) * NN + n0;
#pragma unroll
    for (int v = 0; v < 8; ++v) {
      const int m = v + 8 * hi;
      float r = acc[v] + bias[c0 + m];
      r = r > 0.0f ? r : 0.0f;
      outp[(size_t)m * NN + m15] = r;
    }
  }
}

extern "C" void kernel_launch(void* const* d_in, const int* in_sizes, int n_in,
                              void* d_out, int out_size, void* d_ws, size_t ws_size,
                              hipStream_t stream) {
  (void)in_sizes; (void)n_in; (void)out_size; (void)ws_size;
  const float* x    = (const float*)d_in[0];   // [B, C_IN, N, 1]
  const int*   edge = (const int*)d_in[1];     // [2, B, N, K]
  const float* W    = (const float*)d_in[2];   // [C_OUT, 2*C_IN/GROUPS]
  const float* bias = (const float*)d_in[3];   // [C_OUT]
  float* out = (float*)d_out;                  // [B, C_OUT, N, 1]
  float* xT  = (float*)d_ws;                   // needs B*N*C_IN*4 = 50.3 MB

  dim3 tb(NN / 32, C_IN / 32, BB);
  xpose_kernel<<<tb, dim3(32, 8), 0, stream>>>(x, xT);
  mrconv_kernel<<<BB * NN / NT, 256, 0, stream>>>(xT, edge, W, bias, out);
}